// DeformableDetrMultiscaleDeformableAttention_23235773071473
// MI455X (gfx1250) — compile-verified
//
#include <hip/hip_runtime.h>
#include <hip/hip_bf16.h>
#include <math.h>

// ---------------------------------------------------------------------------
// Deformable DETR multiscale deformable attention, CDNA5 (gfx1250, wave32).
//
// Pipeline (all on `stream`):
//   1) gemm_f32_wmma: value = enc @ W_value + b_value          -> ws
//   2) gemm_f32_wmma: off   = hs  @ W_off   + b_off            -> ws
//   3) gemm_f32_wmma: awlog = hs  @ W_attn  + b_attn           -> ws
//   4) softmax16:     aw    = softmax(awlog) over 16           -> d_out[aw section]
//   5) deform_sample: bilinear gather-accumulate               -> ws (attended)
//   6) gemm_f32_wmma: out   = attended @ W_out + b_out         -> d_out[out section]
//
// GEMMs: V_WMMA_F32_16X16X4_F32 (fp32 matrix core), A tile staged in LDS
// (one workgroup per 16-row strip -> A loaded from HBM exactly once).
// ---------------------------------------------------------------------------

#define D_MODEL 256
#define N_HEADS 8
#define N_LEVELS 4
#define N_POINTS 4
#define DH 32              // D_MODEL / N_HEADS
#define HLP 128            // N_HEADS*N_LEVELS*N_POINTS
#define KMAX 256           // K for all GEMMs here

typedef __attribute__((ext_vector_type(2))) float v2f;
typedef __attribute__((ext_vector_type(8))) float v8f;

// ---------------------------------------------------------------------------
// C[M,N] = A[M,K] @ W[K,N] + bias[N]   (row-major, fp32, WMMA 16x16x4)
//
// One 256-thread block (8 waves) per 16-row M strip. A[16][K] is staged in
// LDS once (coalesced float4), then each wave computes N/16/8 tiles of 16x16:
//   - A fragment from LDS (ds_load_b64 per k-step, shared data across waves)
//   - B fragment streamed from global (per-wave columns)
//
// Fragment layouts (ISA 7.12.2, wave32, 32-bit):
//   A 16x4: lane%16 = M row; VGPR0 = K(half0)/K+2(half1); VGPR1 = K+1/K+3
//   B 4x16: lane%16 = N col; VGPR0 = K row(half0)/K+2(half1); VGPR1 = K+1/K+3
//   C 16x16: VGPR v: row = half*8 + v, col = lane%16
// ---------------------------------------------------------------------------
__global__ __launch_bounds__(256)
void gemm_f32_wmma(const float* __restrict__ A, const float* __restrict__ W,
                   const float* __restrict__ bias, float* __restrict__ C,
                   int M, int N, int K)
{
    __shared__ float As[16 * KMAX];         // 16 KB (of 320 KB WGP LDS)

    const int tid = threadIdx.x;
    const int m0  = blockIdx.x << 4;

    // --- stage A[16][K] into LDS (tail rows clamped; stores are masked) ---
    const int vecs = (16 * K) >> 2;         // float4 count
    for (int v = tid; v < vecs; v += 256) {
        const int idx = v << 2;
        const int r   = idx / K;
        const int c   = idx - r * K;
        int row = m0 + r; if (row >= M) row = M - 1;
        const float4 d = *(const float4*)(A + (size_t)row * K + c);
        *(float4*)(&As[idx]) = d;
    }
    __syncthreads();

    const int wib  = tid >> 5;              // wave in block: 0..7
    const int lane = tid & 31;
    const int half = lane >> 4;             // 0 or 1
    const int l16  = lane & 15;
    const int tpw  = N >> 7;                // (N/16)/8 tiles per wave (1 or 2)

    const float* __restrict__ arow = &As[l16 * K];

    for (int i = 0; i < tpw; ++i) {
        const int n0  = (wib + (i << 3)) << 4;
        const int col = n0 + l16;

        v8f acc = {};
        #pragma unroll 4
        for (int k = 0; k < K; k += 4) {
            const int kh = k + (half << 1);
            v2f a;
            a.x = arow[kh];                 // ds_load_b64 (contiguous pair)
            a.y = arow[kh + 1];
            const float* __restrict__ bp = W + kh * N + col;   // 32-bit index ok
            v2f b;
            b.x = bp[0];
            b.y = bp[N];
            acc = __builtin_amdgcn_wmma_f32_16x16x4_f32(
                /*neg_a=*/false, a, /*neg_b=*/false, b,
                /*c_mod=*/(short)0, acc, /*reuse_a=*/false, /*reuse_b=*/false);
        }

        const float bb = bias[col];
        #pragma unroll
        for (int v = 0; v < 8; ++v) {
            const int row = m0 + (half << 3) + v;
            if (row < M) C[(size_t)row * N + col] = acc[v] + bb;
        }
    }
}

// ---------------------------------------------------------------------------
// Softmax over 16 contiguous logits per (b,q,h).
// logits: (B*Q, 128) with feature = h*16 + l*4 + p; out same layout.
// ---------------------------------------------------------------------------
__global__ __launch_bounds__(256)
void softmax16(const float* __restrict__ logits, float* __restrict__ out, int total)
{
    const int i = blockIdx.x * blockDim.x + threadIdx.x;
    if (i >= total) return;
    const float* __restrict__ p = logits + (size_t)i * 16;
    float v[16];
    float m = -3.4e38f;
    #pragma unroll
    for (int t = 0; t < 16; ++t) { v[t] = p[t]; m = fmaxf(m, v[t]); }
    float s = 0.f;
    #pragma unroll
    for (int t = 0; t < 16; ++t) { v[t] = __expf(v[t] - m); s += v[t]; }
    const float inv = 1.f / s;
    float* __restrict__ o = out + (size_t)i * 16;
    #pragma unroll
    for (int t = 0; t < 16; ++t) o[t] = v[t] * inv;
}

// ---------------------------------------------------------------------------
// Deformable sampling. One wave per (b,q,h); lane = channel (Dh=32 == wave32).
// value:    (B, S, 256)  feature = h*32 + ch   (27 MB -> L2-resident, 192 MB L2)
// off:      (B*Q, 256)   feature = h*32 + l*8 + p*2 + xy
// aw:       (B*Q, 128)   feature = h*16 + l*4 + p   (post-softmax, from d_out)
// refp:     (B*Q, L, 2)
// attended: (B*Q, 256)   feature = h*32 + ch
// All per-point scalars are wave-uniform; corner gathers are contiguous
// 32-lane b32 loads. 32-bit index math throughout (buffers < 2^31 elements).
// ---------------------------------------------------------------------------
__global__ __launch_bounds__(256)
void deform_sample(const float* __restrict__ value, const float* __restrict__ off,
                   const float* __restrict__ aw, const float* __restrict__ refp,
                   const int* __restrict__ shapes, const int* __restrict__ lstart,
                   float* __restrict__ attended, int BQ, int S)
{
    const int wave = blockIdx.x * (blockDim.x >> 5) + (threadIdx.x >> 5);
    const int lane = threadIdx.x & 31;
    if (wave >= BQ * N_HEADS) return;

    const int h  = wave & (N_HEADS - 1);
    const int bq = wave >> 3;
    const int b  = bq / (BQ / 2);           // B == 2

    const float* __restrict__ op = off  + bq * D_MODEL + h * 32;
    const float* __restrict__ ap = aw   + bq * HLP     + h * 16;
    const float* __restrict__ rp = refp + bq * (N_LEVELS * 2);
    const float* __restrict__ vb = value + (b * S) * D_MODEL + h * DH + lane;

    float accv = 0.f;

    #pragma unroll
    for (int l = 0; l < N_LEVELS; ++l) {
        const int   hh = shapes[l * 2 + 0];
        const int   ww = shapes[l * 2 + 1];
        const int   st = lstart[l];
        const float rx = rp[l * 2 + 0];
        const float ry = rp[l * 2 + 1];
        const float fw = (float)ww, fh = (float)hh;

        #pragma unroll
        for (int p = 0; p < N_POINTS; ++p) {
            const float ox = op[l * 8 + p * 2 + 0];
            const float oy = op[l * 8 + p * 2 + 1];
            // loc = ref + off/norm ; x = loc_x*w - 0.5, y = loc_y*h - 0.5
            const float x = (rx + ox / fw) * fw - 0.5f;
            const float y = (ry + oy / fh) * fh - 0.5f;
            const float x0f = floorf(x), y0f = floorf(y);
            const int   x0 = (int)x0f, y0 = (int)y0f;
            const int   x1 = x0 + 1,  y1 = y0 + 1;
            const float wx1 = x - x0f, wx0 = 1.f - wx1;
            const float wy1 = y - y0f, wy0 = 1.f - wy1;

            float s = 0.f;
            // 4 corners; (xi,yi) wave-uniform -> coalesced 32-lane loads
            if (x0 >= 0 && x0 < ww && y0 >= 0 && y0 < hh)
                s += wx0 * wy0 * vb[(st + y0 * ww + x0) * D_MODEL];
            if (x1 >= 0 && x1 < ww && y0 >= 0 && y0 < hh)
                s += wx1 * wy0 * vb[(st + y0 * ww + x1) * D_MODEL];
            if (x0 >= 0 && x0 < ww && y1 >= 0 && y1 < hh)
                s += wx0 * wy1 * vb[(st + y1 * ww + x0) * D_MODEL];
            if (x1 >= 0 && x1 < ww && y1 >= 0 && y1 < hh)
                s += wx1 * wy1 * vb[(st + y1 * ww + x1) * D_MODEL];

            accv += ap[l * 4 + p] * s;
        }
    }
    attended[bq * D_MODEL + h * DH + lane] = accv;
}

// ---------------------------------------------------------------------------
// Launch
// ---------------------------------------------------------------------------
extern "C" void kernel_launch(void* const* d_in, const int* in_sizes, int n_in,
                              void* d_out, int out_size, void* d_ws, size_t ws_size,
                              hipStream_t stream)
{
    const float* hs     = (const float*)d_in[0];   // (B,Q,256)
    const float* enc    = (const float*)d_in[1];   // (B,S,256)
    const float* refp   = (const float*)d_in[2];   // (B,Q,L,2)
    const float* Wv     = (const float*)d_in[3];
    const float* bv     = (const float*)d_in[4];
    const float* Woff   = (const float*)d_in[5];
    const float* boff   = (const float*)d_in[6];
    const float* Wattn  = (const float*)d_in[7];
    const float* battn  = (const float*)d_in[8];
    const float* Wout   = (const float*)d_in[9];
    const float* bout   = (const float*)d_in[10];
    const int*   shapes = (const int*)d_in[11];    // (L,2)
    const int*   lstart = (const int*)d_in[12];    // (L,)

    const int BQ = in_sizes[0] / D_MODEL;          // B*Q = 26588
    const int BS = in_sizes[1] / D_MODEL;          // B*S = 26588
    const int S  = BS / 2;                         // B == 2

    float* out_sec = (float*)d_out;                          // (B*Q, 256)
    float* aw_sec  = (float*)d_out + (size_t)BQ * D_MODEL;   // (B*Q, 128)

    float* ws       = (float*)d_ws;
    float* value    = ws;                                    // BS*256
    float* off      = value + (size_t)BS * D_MODEL;          // BQ*256
    float* awlog    = off + (size_t)BQ * D_MODEL;            // BQ*128
    float* attended = awlog + (size_t)BQ * HLP;              // BQ*256

    const int mb_s = (BS + 15) / 16;               // M-strip blocks
    const int mb_q = (BQ + 15) / 16;

    // 1) value = enc @ W_value + b_value
    gemm_f32_wmma<<<mb_s, 256, 0, stream>>>(enc, Wv, bv, value, BS, D_MODEL, D_MODEL);
    // 2) off = hs @ W_off + b_off
    gemm_f32_wmma<<<mb_q, 256, 0, stream>>>(hs, Woff, boff, off, BQ, D_MODEL, D_MODEL);
    // 3) aw logits = hs @ W_attn + b_attn
    gemm_f32_wmma<<<mb_q, 256, 0, stream>>>(hs, Wattn, battn, awlog, BQ, HLP, D_MODEL);
    // 4) aw = softmax16(awlog) -> directly into d_out aw section
    {
        const int total = BQ * N_HEADS;
        softmax16<<<(total + 255) / 256, 256, 0, stream>>>(awlog, aw_sec, total);
    }
    // 5) deformable bilinear sampling -> attended
    {
        const long long waves = (long long)BQ * N_HEADS;
        const int blocks = (int)((waves + 7) / 8);
        deform_sample<<<blocks, 256, 0, stream>>>(
            value, off, aw_sec, refp, shapes, lstart, attended, BQ, S);
    }
    // 6) out = attended @ W_out + b_out -> d_out out section
    gemm_f32_wmma<<<mb_q, 256, 0, stream>>>(attended, Wout, bout, out_sec, BQ, D_MODEL, D_MODEL);
}